// LiteGTNet_65420941853359
// MI455X (gfx1250) — compile-verified
//
#include <hip/hip_runtime.h>

#define N_NODES 100000
#define N_EDGES 3200000
#define HID     128
#define POSD    8
#define L0D     64
#define L1D     32
#define NC      7
#define TOPK    800
#define CAP     4096

typedef float v2f __attribute__((ext_vector_type(2)));
typedef float v8f __attribute__((ext_vector_type(8)));

// ---------------------------------------------------------------------------
// Fused node pipeline: h = emb[tok] + pos@lapW + lapb  (kept in LDS only),
// then MLPReadout 128 -> 64 -> 32 -> 7, all via V_WMMA_F32_16X16X4_F32.
// Block = 256 threads = 8 wave32s, one 16-row tile per block.
// ---------------------------------------------------------------------------
__global__ __launch_bounds__(256) void node_mlp_kernel(
    const int* __restrict__ tokens,
    const float* __restrict__ pos,
    const float* __restrict__ emb,
    const float* __restrict__ lapW,
    const float* __restrict__ lapb,
    const float* __restrict__ W0, const float* __restrict__ b0,
    const float* __restrict__ W1, const float* __restrict__ b1,
    const float* __restrict__ W2, const float* __restrict__ b2,
    float* __restrict__ outh)
{
  __shared__ float hs [16 * HID];
  __shared__ float x0s[16 * L0D];
  __shared__ float x1s[16 * L1D];

  const int tid  = threadIdx.x;
  const int wv   = tid >> 5;      // wave32: 8 waves per block
  const int lane = tid & 31;
  const int mn   = lane & 15;     // M row (A/C) or N col (B/C)
  const int hi   = lane >> 4;     // half-wave selector (K / M split)
  const int r0   = blockIdx.x * 16;

  // ---- Stage 0: h tile (all 8 waves, one 16-col tile each) ----------------
  {
    const int c0 = wv * 16;
    v8f acc = {0.f,0.f,0.f,0.f,0.f,0.f,0.f,0.f};
#pragma unroll
    for (int kc = 0; kc < POSD / 4; ++kc) {
      const int kb = kc * 4 + hi * 2;           // A: lanes 0-15 K{0,1}, 16-31 K{2,3}
      v2f a, b;
      a.x = pos[(r0 + mn) * POSD + kb];
      a.y = pos[(r0 + mn) * POSD + kb + 1];
      b.x = lapW[(kb    ) * HID + c0 + mn];
      b.y = lapW[(kb + 1) * HID + c0 + mn];
      acc = __builtin_amdgcn_wmma_f32_16x16x4_f32(false, a, false, b,
                                                  (short)0, acc, false, false);
    }
#pragma unroll
    for (int v = 0; v < 8; ++v) {               // C/D: VGPR v -> M = v + 8*hi
      const int M = v + hi * 8;
      const int t = tokens[r0 + M];
      hs[M * HID + c0 + mn] = acc[v] + emb[t * HID + c0 + mn] + lapb[c0 + mn];
    }
  }
  __syncthreads();

  // ---- Stage 1: x0 = relu(h @ W0 + b0)  [16 x 64], waves 0-3 --------------
  if (wv < 4) {
    const int c0 = wv * 16;
    v8f acc = {0.f,0.f,0.f,0.f,0.f,0.f,0.f,0.f};
#pragma unroll 4
    for (int kc = 0; kc < HID / 4; ++kc) {
      const int kb = kc * 4 + hi * 2;
      v2f a, b;
      a.x = hs[mn * HID + kb];
      a.y = hs[mn * HID + kb + 1];
      b.x = W0[(kb    ) * L0D + c0 + mn];
      b.y = W0[(kb + 1) * L0D + c0 + mn];
      acc = __builtin_amdgcn_wmma_f32_16x16x4_f32(false, a, false, b,
                                                  (short)0, acc, false, false);
    }
#pragma unroll
    for (int v = 0; v < 8; ++v) {
      const int M = v + hi * 8;
      x0s[M * L0D + c0 + mn] = fmaxf(acc[v] + b0[c0 + mn], 0.f);
    }
  }
  __syncthreads();

  // ---- Stage 2: x1 = relu(x0 @ W1 + b1)  [16 x 32], waves 0-1 -------------
  if (wv < 2) {
    const int c0 = wv * 16;
    v8f acc = {0.f,0.f,0.f,0.f,0.f,0.f,0.f,0.f};
#pragma unroll
    for (int kc = 0; kc < L0D / 4; ++kc) {
      const int kb = kc * 4 + hi * 2;
      v2f a, b;
      a.x = x0s[mn * L0D + kb];
      a.y = x0s[mn * L0D + kb + 1];
      b.x = W1[(kb    ) * L1D + c0 + mn];
      b.y = W1[(kb + 1) * L1D + c0 + mn];
      acc = __builtin_amdgcn_wmma_f32_16x16x4_f32(false, a, false, b,
                                                  (short)0, acc, false, false);
    }
#pragma unroll
    for (int v = 0; v < 8; ++v) {
      const int M = v + hi * 8;
      x1s[M * L1D + c0 + mn] = fmaxf(acc[v] + b1[c0 + mn], 0.f);
    }
  }
  __syncthreads();

  // ---- Stage 3: out = x1 @ W2 + b2  [16 x 7], wave 0 ----------------------
  if (wv == 0) {
    v8f acc = {0.f,0.f,0.f,0.f,0.f,0.f,0.f,0.f};
#pragma unroll
    for (int kc = 0; kc < L1D / 4; ++kc) {
      const int kb = kc * 4 + hi * 2;
      v2f a, b;
      a.x = x1s[mn * L1D + kb];
      a.y = x1s[mn * L1D + kb + 1];
      b.x = (mn < NC) ? W2[(kb    ) * NC + mn] : 0.f;
      b.y = (mn < NC) ? W2[(kb + 1) * NC + mn] : 0.f;
      acc = __builtin_amdgcn_wmma_f32_16x16x4_f32(false, a, false, b,
                                                  (short)0, acc, false, false);
    }
#pragma unroll
    for (int v = 0; v < 8; ++v) {
      const int M = v + hi * 8;
      if (mn < NC) outh[(r0 + M) * NC + mn] = acc[v] + b2[mn];
    }
  }
}

// ---------------------------------------------------------------------------
// Edge pass 1: per-dst degree / sum / max (atomics into zeroed workspace).
// e >= 0, so float-max == uint-max on bit patterns.
// ---------------------------------------------------------------------------
__global__ void edge_deg_kernel(const int* __restrict__ dst,
                                const float* __restrict__ e,
                                float* __restrict__ deg,
                                float* __restrict__ sum,
                                unsigned* __restrict__ smax, int nE)
{
  for (int i = blockIdx.x * blockDim.x + threadIdx.x; i < nE;
       i += gridDim.x * blockDim.x) {
    const int   d  = dst[i];
    const float ev = e[i];
    atomicAdd(&deg[d], 1.0f);
    atomicAdd(&sum[d], ev);
    atomicMax(&smax[d], __float_as_uint(ev));
  }
}

// ---------------------------------------------------------------------------
// Node stats: jaccard_M output, inv_deg for score kernel, monotone sort key.
// ---------------------------------------------------------------------------
__global__ void node_stat_kernel(const float* __restrict__ deg,
                                 const float* __restrict__ sum,
                                 const unsigned* __restrict__ smax,
                                 float* __restrict__ inv,
                                 unsigned* __restrict__ keys,
                                 float* __restrict__ outjac, int n)
{
  const int i = blockIdx.x * blockDim.x + threadIdx.x;
  if (i >= n) return;
  const float d    = deg[i];
  const float dsaf = fmaxf(d, 1.0f);
  const float iv   = 1.0f / dsaf;
  inv[i] = iv;
  const float mx  = __uint_as_float(smax[i]);
  const float jac = (d > 0.0f) ? (mx - sum[i] * iv - __logf(dsaf))
                               : -__builtin_inff();
  outjac[i] = jac;
  const unsigned u = __float_as_uint(jac);
  keys[i] = (u & 0x80000000u) ? ~u : (u | 0x80000000u);   // larger f -> larger key
}

// ---------------------------------------------------------------------------
// Edge pass 2: score[e, h] = inv_deg[dst[e]] for 4 score heads (b128 stores).
// ---------------------------------------------------------------------------
__global__ void score_kernel(const int* __restrict__ dst,
                             const float* __restrict__ inv,
                             float4* __restrict__ score, int nE)
{
  const int i = blockIdx.x * blockDim.x + threadIdx.x;
  if (i >= nE) return;
  const float v = inv[dst[i]];
  float4 v4; v4.x = v; v4.y = v; v4.z = v; v4.w = v;
  score[i] = v4;
}

// ---------------------------------------------------------------------------
// Radix-select top-K: state = {prefix, mask, selected_gt_count, gather_count}
// ---------------------------------------------------------------------------
__global__ void hist_kernel(const unsigned* __restrict__ keys,
                            const unsigned* __restrict__ state,
                            unsigned* __restrict__ hist, int n, int shift)
{
  __shared__ unsigned lh[256];
  for (int i = threadIdx.x; i < 256; i += blockDim.x) lh[i] = 0u;
  __syncthreads();
  const unsigned prefix = state[0], mask = state[1];
  for (int i = blockIdx.x * blockDim.x + threadIdx.x; i < n;
       i += gridDim.x * blockDim.x) {
    const unsigned k = keys[i];
    if ((k & mask) == prefix) atomicAdd(&lh[(k >> shift) & 0xFFu], 1u);
  }
  __syncthreads();
  for (int i = threadIdx.x; i < 256; i += blockDim.x)
    if (lh[i]) atomicAdd(&hist[i], lh[i]);
}

__global__ void scan_kernel(unsigned* __restrict__ state,
                            unsigned* __restrict__ hist, int shift, int K)
{
  __shared__ unsigned h[256];
  h[threadIdx.x] = hist[threadIdx.x];
  __syncthreads();
  if (threadIdx.x == 0) {
    const int remaining = K - (int)state[2];
    unsigned cum = 0; int chosen = 0; unsigned above = 0;
    for (int b = 255; b >= 0; --b) {
      const unsigned c = h[b];
      if ((int)(cum + c) >= remaining) { chosen = b; above = cum; break; }
      cum += c;
    }
    state[0] |= ((unsigned)chosen) << shift;
    state[1] |= 0xFFu << shift;
    state[2] += above;
  }
  __syncthreads();
  hist[threadIdx.x] = 0u;   // ready for next pass
}

__global__ void gather_kernel(const unsigned* __restrict__ keys,
                              unsigned* __restrict__ state,
                              unsigned* __restrict__ gkey,
                              unsigned* __restrict__ gidx, int n)
{
  const unsigned T = state[0];   // exact threshold key after 4 passes
  for (int i = blockIdx.x * blockDim.x + threadIdx.x; i < n;
       i += gridDim.x * blockDim.x) {
    const unsigned k = keys[i];
    if (k >= T) {
      const unsigned p = atomicAdd(&state[3], 1u);
      if (p < CAP) { gkey[p] = k; gidx[p] = (unsigned)i; }
    }
  }
}

// Single-block bitonic sort of <=CAP candidates: key desc, index asc on ties.
__global__ __launch_bounds__(1024) void sort_out_kernel(
    const unsigned* __restrict__ state,
    const unsigned* __restrict__ gkey,
    const unsigned* __restrict__ gidx,
    int* __restrict__ outidx, int K)
{
  __shared__ unsigned sk[CAP];
  __shared__ unsigned si[CAP];
  unsigned n = state[3]; if (n > CAP) n = CAP;
  for (int i = threadIdx.x; i < CAP; i += 1024) {
    if (i < (int)n) { sk[i] = gkey[i]; si[i] = gidx[i]; }
    else            { sk[i] = 0u;      si[i] = 0x7FFFFFFFu; }
  }
  __syncthreads();
  for (int kk = 2; kk <= CAP; kk <<= 1) {
    for (int j = kk >> 1; j > 0; j >>= 1) {
      for (int i = threadIdx.x; i < CAP; i += 1024) {
        const int ixj = i ^ j;
        if (ixj > i) {
          const unsigned ka = sk[i], kb = sk[ixj];
          const unsigned ia = si[i], ib = si[ixj];
          const bool b_before_a = (kb > ka) || (kb == ka && ib < ia);
          const bool a_before_b = (ka > kb) || (ka == kb && ia < ib);
          const bool sw = ((i & kk) == 0) ? b_before_a : a_before_b;
          if (sw) { sk[i] = kb; sk[ixj] = ka; si[i] = ib; si[ixj] = ia; }
        }
      }
      __syncthreads();
    }
  }
  for (int i = threadIdx.x; i < K; i += 1024) outidx[i] = (int)si[i];
}

// ---------------------------------------------------------------------------
extern "C" void kernel_launch(void* const* d_in, const int* in_sizes, int n_in,
                              void* d_out, int out_size, void* d_ws, size_t ws_size,
                              hipStream_t stream)
{
  (void)in_sizes; (void)n_in; (void)out_size; (void)ws_size;

  const int*   tokens = (const int*)  d_in[0];
  /* src = d_in[1] unused by reference outputs */
  const int*   dst    = (const int*)  d_in[2];
  const float* e      = (const float*)d_in[3];
  const float* pos    = (const float*)d_in[4];
  const float* emb    = (const float*)d_in[5];
  const float* lapW   = (const float*)d_in[6];
  const float* lapb   = (const float*)d_in[7];
  const float* W0     = (const float*)d_in[8];
  const float* b0     = (const float*)d_in[9];
  const float* W1     = (const float*)d_in[10];
  const float* b1     = (const float*)d_in[11];
  const float* W2     = (const float*)d_in[12];
  const float* b2     = (const float*)d_in[13];

  // d_out: h_out [N,7] | score [E,4] | jaccard_M [N] | top_idx [K]
  float* outh   = (float*)d_out;
  float* outsc  = outh  + (size_t)N_NODES * NC;
  float* outjac = outsc + (size_t)N_EDGES * 4;
  int*   outidx = (int*)(outjac + N_NODES);

  // workspace layout (all 4-byte units)
  float*    deg   = (float*)d_ws;
  float*    sum   = deg + N_NODES;
  unsigned* smax  = (unsigned*)(sum + N_NODES);
  float*    inv   = (float*)(smax + N_NODES);
  unsigned* keys  = (unsigned*)(inv + N_NODES);
  unsigned* state = keys + N_NODES;        // 8
  unsigned* hist  = state + 8;             // 256
  unsigned* gkey  = hist + 256;            // CAP
  unsigned* gidx  = gkey + CAP;            // CAP

  hipMemsetAsync(deg,   0, (size_t)3 * N_NODES * sizeof(float), stream);
  hipMemsetAsync(state, 0, (8 + 256) * sizeof(unsigned), stream);

  node_mlp_kernel<<<N_NODES / 16, 256, 0, stream>>>(
      tokens, pos, emb, lapW, lapb, W0, b0, W1, b1, W2, b2, outh);

  edge_deg_kernel<<<2048, 256, 0, stream>>>(dst, e, deg, sum, smax, N_EDGES);

  node_stat_kernel<<<(N_NODES + 255) / 256, 256, 0, stream>>>(
      deg, sum, smax, inv, keys, outjac, N_NODES);

  score_kernel<<<(N_EDGES + 255) / 256, 256, 0, stream>>>(
      dst, inv, (float4*)outsc, N_EDGES);

  for (int p = 0; p < 4; ++p) {
    const int shift = 24 - 8 * p;
    hist_kernel<<<512, 256, 0, stream>>>(keys, state, hist, N_NODES, shift);
    scan_kernel<<<1, 256, 0, stream>>>(state, hist, shift, TOPK);
  }
  gather_kernel<<<512, 256, 0, stream>>>(keys, state, gkey, gidx, N_NODES);
  sort_out_kernel<<<1, 1024, 0, stream>>>(state, gkey, gidx, outidx, TOPK);
}